// DS_Block_61117384622260
// MI455X (gfx1250) — compile-verified
//
#include <hip/hip_runtime.h>
#include <math.h>

// MI455X / gfx1250, wave32. All GEMM-shaped work goes through
// V_WMMA_F32_16X16X4_F32. Laplacian is factored (rank-1 + I) to O(B*N*C).

#define B_  16
#define N_  2000
#define C_  128
#define KNN 9
#define NT_ 125   // N_/16

typedef float v2f __attribute__((ext_vector_type(2)));
typedef float v8f __attribute__((ext_vector_type(8)));

__device__ __forceinline__ v8f wmma_f32_4(v2f a, v2f b, v8f c) {
  // 8 args: (neg_a, A, neg_b, B, c_mod, C, reuse_a, reuse_b)
  return __builtin_amdgcn_wmma_f32_16x16x4_f32(false, a, false, b, (short)0, c,
                                               false, false);
}

// ---------------- workspace layout (floats) ----------------
static constexpr size_t OFF_H    = 0;                               // [B][C][N]
static constexpr size_t OFF_SQ   = OFF_H    + (size_t)B_*C_*N_;     // [B][N]
static constexpr size_t OFF_IDX  = OFF_SQ   + (size_t)B_*N_;        // [B][N][K] ints
static constexpr size_t OFF_H1   = OFF_IDX  + (size_t)B_*N_*KNN;    // [B][C][N]
static constexpr size_t OFF_WN   = OFF_H1   + (size_t)B_*C_*N_;     // [B][N]
static constexpr size_t OFF_DM   = OFF_WN   + (size_t)B_*N_;        // [B][N]
static constexpr size_t OFF_COEF = OFF_DM   + (size_t)B_*N_;        // [B][N]
static constexpr size_t OFF_WSUM = OFF_COEF + (size_t)B_*N_;        // [B]
static constexpr size_t OFF_T1   = OFF_WSUM + 16;                   // [B][C]
static constexpr size_t OFF_H2   = OFF_T1   + (size_t)B_*C_;        // [B][C][N]
static constexpr size_t OFF_Q    = OFF_H2   + (size_t)B_*C_*N_;     // [B][32][N]
static constexpr size_t OFF_KK   = OFF_Q    + (size_t)B_*32*N_;     // [B][32][N]
static constexpr size_t OFF_V    = OFF_KK   + (size_t)B_*32*N_;     // [B][C][N]
static constexpr size_t OFF_AO   = OFF_V    + (size_t)B_*C_*N_;     // [B][C][N]
static constexpr size_t OFF_H3   = OFF_AO   + (size_t)B_*C_*N_;     // [B][C][N]
static constexpr size_t OFF_T2   = OFF_H3   + (size_t)B_*C_*N_;     // [B][C]

// ---------------- embed: h = relu(W0 x + b0); sq = sum_c h^2 ----------------
__global__ void k_embed(const float* __restrict__ x, const float* __restrict__ W0,
                        const float* __restrict__ b0, float* __restrict__ h,
                        float* __restrict__ sq) {
  int p = blockIdx.x;                 // b*N_ + n
  int b = p / N_, n = p - b * N_;
  int c = threadIdx.x;                // 128 threads
  float x0 = x[(b*4+0)*N_ + n];
  float x1 = x[(b*4+1)*N_ + n];
  float x2 = x[(b*4+2)*N_ + n];
  float x3 = x[(b*4+3)*N_ + n];
  float v = W0[c*4+0]*x0 + W0[c*4+1]*x1 + W0[c*4+2]*x2 + W0[c*4+3]*x3 + b0[c];
  v = fmaxf(v, 0.0f);
  h[((size_t)b*C_ + c)*N_ + n] = v;
  __shared__ float red[128];
  red[c] = v * v;
  __syncthreads();
  for (int s = 64; s > 0; s >>= 1) {
    if (c < s) red[c] += red[c + s];
    __syncthreads();
  }
  if (c == 0) sq[p] = red[0];
}

// ---------------- kNN: WMMA gram tiles + fused per-row top-9 ----------------
__global__ void __launch_bounds__(32)
k_knn(const float* __restrict__ h, const float* __restrict__ sq,
      int* __restrict__ idx) {
  int blk = blockIdx.x;
  int b = blk / NT_;
  int n0 = (blk - b*NT_) * 16;
  int lane = threadIdx.x;
  int r = lane & 15, hi = lane >> 4;
  const float* hb  = h  + (size_t)b*C_*N_;
  const float* sqb = sq + (size_t)b*N_;

  // A operand: 16 query rows, preloaded for all 32 K-steps (K=128)
  v2f a[32];
#pragma unroll
  for (int s = 0; s < 32; ++s) {
    int k = 4*s + 2*hi;
    a[s].x = hb[(size_t)k    *N_ + n0 + r];
    a[s].y = hb[(size_t)(k+1)*N_ + n0 + r];
  }
  float sqn = sqb[n0 + r];

  float tv[KNN]; int ti[KNN];
#pragma unroll
  for (int t = 0; t < KNN; ++t) { tv[t] = -3.4e38f; ti[t] = 0; }

  __shared__ __align__(16) float stile[256];

  for (int mt = 0; mt < NT_; ++mt) {
    int m0 = mt * 16;
    v8f acc = {0.f,0.f,0.f,0.f,0.f,0.f,0.f,0.f};
#pragma unroll
    for (int s = 0; s < 32; ++s) {
      int k = 4*s + 2*hi;
      v2f bb;
      bb.x = hb[(size_t)k    *N_ + m0 + r];
      bb.y = hb[(size_t)(k+1)*N_ + m0 + r];
      acc = wmma_f32_4(a[s], bb, acc);
    }
#pragma unroll
    for (int i = 0; i < 8; ++i) stile[(i + 8*hi)*16 + r] = acc[i];
    __syncthreads();
    if (lane < 16) {                 // lane owns one query row
      const float* prow = &stile[lane*16];
#pragma unroll
      for (int col = 0; col < 16; ++col) {
        int m = m0 + col;
        float d = 2.0f*prow[col] - sqn - sqb[m];
        if (d > tv[KNN-1]) {         // register-resident sorted insert (stable)
          float cv = d; int ci = m;
#pragma unroll
          for (int t = 0; t < KNN; ++t) {
            bool g = cv > tv[t];
            float nv = g ? cv : tv[t]; int ni = g ? ci : ti[t];
            float ov = tv[t];          int oi = ti[t];
            cv = g ? ov : cv;          ci = g ? oi : ci;
            tv[t] = nv;                ti[t] = ni;
          }
        }
      }
    }
    __syncthreads();
  }
  if (lane < 16) {
    int n = n0 + lane;
#pragma unroll
    for (int t = 0; t < KNN; ++t) idx[((size_t)b*N_ + n)*KNN + t] = ti[t];
  }
}

// ------- SCE: gather graph feature into LDS panel, two WMMA GEMMs -----------
// Bp1: [48 cols = j*16+pt][768 k = c*3+t]  (172KB total dyn LDS, WGP has 320KB)
#define SCE_LDS_BYTES ((48*768 + 16*384) * sizeof(float))
__global__ void __launch_bounds__(256)
k_sce(const float* __restrict__ h, const int* __restrict__ idx,
      const float* __restrict__ W1, const float* __restrict__ b1,
      const float* __restrict__ W2, const float* __restrict__ b2,
      float* __restrict__ h1) {
  extern __shared__ float smem[];
  float* Bp1 = smem;              // [48][768]
  float* Bp2 = smem + 48*768;     // [16][384]

  int blk = blockIdx.x;
  int b = blk / NT_;
  int n0 = (blk - b*NT_) * 16;
  int tid = threadIdx.x;
  const float* hb = h + (size_t)b*C_*N_;

  // phase 1: build gathered B panel  g3[c,n,j,t] -> Bp1[j*16+pt][c*3+t]
  for (int e = tid; e < 48*768; e += 256) {
    int col = e / 768, k = e - col*768;
    int j = col >> 4, pt = col & 15;
    int c = k / 3, t = k - c*3;
    int n = n0 + pt;
    float val;
    if (c < C_) {
      val = hb[(size_t)c*N_ + n];                       // center
    } else {
      int cc = c - C_;
      int m = idx[((size_t)b*N_ + n)*KNN + j*3 + t];
      val = hb[(size_t)cc*N_ + n] - hb[(size_t)cc*N_ + m];  // center - feat
    }
    Bp1[e] = val;
  }
  __syncthreads();

  int wave = tid >> 5;            // one wave per 16-output-channel tile
  int lane = tid & 31;
  int r = lane & 15, hi = lane >> 4;

  const v8f vz = {0.f,0.f,0.f,0.f,0.f,0.f,0.f,0.f};
  v8f acc[3];
#pragma unroll
  for (int j = 0; j < 3; ++j) acc[j] = vz;

  // phase 2: conv1  [128 out][768 k][48 cols] ; K-loop 192
  const float* W1row = W1 + (size_t)(wave*16 + r)*768;
  for (int s = 0; s < 192; ++s) {
    int k = 4*s + 2*hi;
    v2f aw = *(const v2f*)(W1row + k);
#pragma unroll
    for (int j = 0; j < 3; ++j) {
      v2f bb = *(const v2f*)(Bp1 + (size_t)(j*16 + r)*768 + k);
      acc[j] = wmma_f32_4(aw, bb, acc[j]);
    }
  }

  // phase 3: bias+relu -> Bp2[pt][o*3+j]
#pragma unroll
  for (int j = 0; j < 3; ++j) {
#pragma unroll
    for (int i = 0; i < 8; ++i) {
      int o = wave*16 + i + 8*hi;
      Bp2[(size_t)r*384 + o*3 + j] = fmaxf(acc[j][i] + b1[o], 0.0f);
    }
  }
  __syncthreads();

  // phase 4: conv2  [128 out][384 k][16 cols]
  v8f acc2 = vz;
  const float* W2row = W2 + (size_t)(wave*16 + r)*384;
  for (int s = 0; s < 96; ++s) {
    int k = 4*s + 2*hi;
    v2f aw = *(const v2f*)(W2row + k);
    v2f bb = *(const v2f*)(Bp2 + (size_t)r*384 + k);
    acc2 = wmma_f32_4(aw, bb, acc2);
  }
#pragma unroll
  for (int i = 0; i < 8; ++i) {
    int o = wave*16 + i + 8*hi;
    h1[((size_t)b*C_ + o)*N_ + n0 + r] = fmaxf(acc2[i] + b2[o], 0.0f);
  }
}

// ------- weight logits -> wn = relu(tanh(Ww.h1 + bw)) -----------------------
__global__ void k_wlog(const float* __restrict__ h1, const float* __restrict__ Ww,
                       const float* __restrict__ bw, float* __restrict__ wn) {
  int g = blockIdx.x*256 + threadIdx.x;      // B_*N_ exactly
  int b = g / N_, n = g - b*N_;
  const float* p = h1 + (size_t)b*C_*N_ + n;
  float s = bw[0];
  for (int c = 0; c < C_; ++c) s += Ww[c] * p[(size_t)c*N_];
  wn[g] = fmaxf(tanhf(s), 0.0f);
}

__global__ void k_wsum(const float* __restrict__ wn, float* __restrict__ wsum) {
  int b = blockIdx.x;
  __shared__ float red[256];
  float s = 0.f;
  for (int n = threadIdx.x; n < N_; n += 256) s += wn[(size_t)b*N_ + n];
  red[threadIdx.x] = s; __syncthreads();
  for (int st = 128; st > 0; st >>= 1) {
    if (threadIdx.x < st) red[threadIdx.x] += red[threadIdx.x + st];
    __syncthreads();
  }
  if (threadIdx.x == 0) wsum[b] = red[0];
}

// Dm = rsqrt(wn*W + 1); coef = wn*Dm
__global__ void k_coef(const float* __restrict__ wn, const float* __restrict__ wsum,
                       float* __restrict__ dm, float* __restrict__ coef) {
  int g = blockIdx.x*256 + threadIdx.x;
  int b = g / N_;
  float w = wn[g];
  float d = rsqrtf(w * wsum[b] + 1.0f);
  dm[g] = d;
  coef[g] = w * d;
}

// T[b][c] = sum_m coef[b][m] * src[b][c][m]   (rank-1 part of Laplacian agg)
__global__ void k_T(const float* __restrict__ coef, const float* __restrict__ src,
                    float* __restrict__ T) {
  int blk = blockIdx.x;           // b*C_ + c
  int b = blk / C_, c = blk - b*C_;
  __shared__ float red[256];
  const float* s  = src  + ((size_t)b*C_ + c)*N_;
  const float* cf = coef + (size_t)b*N_;
  float acc = 0.f;
  for (int n = threadIdx.x; n < N_; n += 256) acc += cf[n]*s[n];
  red[threadIdx.x] = acc; __syncthreads();
  for (int st = 128; st > 0; st >>= 1) {
    if (threadIdx.x < st) red[threadIdx.x] += red[threadIdx.x + st];
    __syncthreads();
  }
  if (threadIdx.x == 0) T[blk] = red[0];
}

// ------- generic channel-mixing conv: out = res + g*relu(W.in + b) ----------
__global__ void __launch_bounds__(32)
k_conv(const float* __restrict__ W, const float* __restrict__ bias,
       const float* __restrict__ in, float* __restrict__ out,
       const float* __restrict__ residual, const float* __restrict__ gamma,
       int ctiles) {
  int blk = blockIdx.x;
  int ct = blk % ctiles; blk /= ctiles;
  int nt = blk % NT_;    int b = blk / NT_;
  int n0 = nt*16;
  int lane = threadIdx.x, r = lane & 15, hi = lane >> 4;

  const float* Wrow = W + (size_t)(ct*16 + r)*C_;
  v2f a[32];
#pragma unroll
  for (int s = 0; s < 32; ++s) a[s] = *(const v2f*)(Wrow + 4*s + 2*hi);

  const float* inb = in + (size_t)b*C_*N_ + n0 + r;
  v8f acc = {0.f,0.f,0.f,0.f,0.f,0.f,0.f,0.f};
#pragma unroll 4
  for (int s = 0; s < 32; ++s) {
    int k = 4*s + 2*hi;
    v2f bb;
    bb.x = inb[(size_t)k    *N_];
    bb.y = inb[(size_t)(k+1)*N_];
    acc = wmma_f32_4(a[s], bb, acc);
  }
  float g = gamma ? gamma[0] : 1.0f;
  int Cout = ctiles*16;
#pragma unroll
  for (int i = 0; i < 8; ++i) {
    int o = ct*16 + i + 8*hi;
    float v = fmaxf(acc[i] + bias[o], 0.0f);
    size_t oix = ((size_t)b*Cout + o)*N_ + n0 + r;
    if (residual) v = residual[oix] + g*v;
    out[oix] = v;
  }
}

// ------- Laplacian-fused conv: in(c,n) = a1(n)*T[c] + a2(n)*src[c][n] -------
__global__ void __launch_bounds__(32)
k_gcn_conv(const float* __restrict__ W, const float* __restrict__ bias,
           const float* __restrict__ src, const float* __restrict__ wn,
           const float* __restrict__ dm, const float* __restrict__ T,
           float* __restrict__ out, const float* __restrict__ residual) {
  int blk = blockIdx.x;
  int ct = blk & 7; blk >>= 3;
  int nt = blk % NT_; int b = blk / NT_;
  int n0 = nt*16;
  int lane = threadIdx.x, r = lane & 15, hi = lane >> 4;
  int n = n0 + r;

  const float* Wrow = W + (size_t)(ct*16 + r)*C_;
  v2f a[32];
#pragma unroll
  for (int s = 0; s < 32; ++s) a[s] = *(const v2f*)(Wrow + 4*s + 2*hi);

  float dmn = dm[(size_t)b*N_ + n];
  float a1 = dmn * wn[(size_t)b*N_ + n];
  float a2 = dmn * dmn;
  const float* srcb = src + (size_t)b*C_*N_ + n;
  const float* Tb = T + (size_t)b*C_;

  v8f acc = {0.f,0.f,0.f,0.f,0.f,0.f,0.f,0.f};
#pragma unroll 4
  for (int s = 0; s < 32; ++s) {
    int k = 4*s + 2*hi;
    v2f bb;
    bb.x = a1*Tb[k]   + a2*srcb[(size_t)k    *N_];
    bb.y = a1*Tb[k+1] + a2*srcb[(size_t)(k+1)*N_];
    acc = wmma_f32_4(a[s], bb, acc);
  }
#pragma unroll
  for (int i = 0; i < 8; ++i) {
    int o = ct*16 + i + 8*hi;
    float v = fmaxf(acc[i] + bias[o], 0.0f);
    size_t oix = ((size_t)b*C_ + o)*N_ + n;
    if (residual) v += residual[oix];
    out[oix] = v;
  }
}

// ------- flash attention: softmax(q^T k) applied to v, streaming ------------
__global__ void __launch_bounds__(32)
k_attn(const float* __restrict__ q, const float* __restrict__ kk,
       const float* __restrict__ v, float* __restrict__ ao) {
  int blk = blockIdx.x;
  int b = blk / NT_;
  int n0 = (blk - b*NT_)*16;
  int lane = threadIdx.x, r = lane & 15, hi = lane >> 4;

  __shared__ __align__(16) float ptile[256];
  __shared__ float resc[16];
  __shared__ float rmax[16];
  __shared__ float rsum[16];

  const float* qb = q  + (size_t)b*32*N_;
  const float* kb = kk + (size_t)b*32*N_;
  const float* vb = v  + (size_t)b*C_*N_;

  v2f aq[8];                      // Q rows, K=32 -> 8 K-steps
#pragma unroll
  for (int s = 0; s < 8; ++s) {
    int k = 4*s + 2*hi;
    aq[s].x = qb[(size_t)k    *N_ + n0 + r];
    aq[s].y = qb[(size_t)(k+1)*N_ + n0 + r];
  }

  const v8f vz = {0.f,0.f,0.f,0.f,0.f,0.f,0.f,0.f};
  v8f accO[8];
#pragma unroll
  for (int m = 0; m < 8; ++m) accO[m] = vz;

  if (lane < 16) { rmax[lane] = -3.4e38f; rsum[lane] = 0.f; }
  __syncthreads();

  for (int mt = 0; mt < NT_; ++mt) {
    int m0 = mt*16;
    v8f sacc = vz;                // S = Q.K^T tile
#pragma unroll
    for (int s = 0; s < 8; ++s) {
      int k = 4*s + 2*hi;
      v2f bb;
      bb.x = kb[(size_t)k    *N_ + m0 + r];
      bb.y = kb[(size_t)(k+1)*N_ + m0 + r];
      sacc = wmma_f32_4(aq[s], bb, sacc);
    }
#pragma unroll
    for (int i = 0; i < 8; ++i) ptile[(i + 8*hi)*16 + r] = sacc[i];
    __syncthreads();
    if (lane < 16) {              // streaming softmax row stats
      float* prow = &ptile[lane*16];
      float mx = rmax[lane], tmax = mx;
#pragma unroll
      for (int c2 = 0; c2 < 16; ++c2) tmax = fmaxf(tmax, prow[c2]);
      float rs = 0.f;
#pragma unroll
      for (int c2 = 0; c2 < 16; ++c2) { float e = __expf(prow[c2]-tmax); prow[c2]=e; rs += e; }
      float rc = __expf(mx - tmax);
      resc[lane] = rc; rmax[lane] = tmax; rsum[lane] = rsum[lane]*rc + rs;
    }
    __syncthreads();
    float rc = resc[r];           // rescale running accumulators (col = n)
#pragma unroll
    for (int m = 0; m < 8; ++m)
#pragma unroll
      for (int i = 0; i < 8; ++i) accO[m][i] *= rc;
    // O += V . P^T
#pragma unroll
    for (int s4 = 0; s4 < 4; ++s4) {
      int km = 4*s4 + 2*hi;
      v2f bp = *(const v2f*)(ptile + r*16 + km);     // P[n=col][m=k]
#pragma unroll
      for (int m = 0; m < 8; ++m) {
        v2f av = *(const v2f*)(vb + (size_t)(m*16 + r)*N_ + m0 + km);
        accO[m] = wmma_f32_4(av, bp, accO[m]);
      }
    }
    __syncthreads();
  }
  float inv = 1.0f / rsum[r];
#pragma unroll
  for (int m = 0; m < 8; ++m)
#pragma unroll
    for (int i = 0; i < 8; ++i)
      ao[((size_t)b*C_ + m*16 + i + 8*hi)*N_ + n0 + r] = accO[m][i] * inv;
}

// ----------------------------------------------------------------------------
extern "C" void kernel_launch(void* const* d_in, const int* in_sizes, int n_in,
                              void* d_out, int out_size, void* d_ws, size_t ws_size,
                              hipStream_t stream) {
  (void)in_sizes; (void)n_in; (void)out_size; (void)ws_size;
  const float* x   = (const float*)d_in[0];
  const float* W0  = (const float*)d_in[1];
  const float* b0  = (const float*)d_in[2];
  const float* W1  = (const float*)d_in[3];
  const float* b1  = (const float*)d_in[4];
  const float* W2  = (const float*)d_in[5];
  const float* b2  = (const float*)d_in[6];
  const float* Ww  = (const float*)d_in[7];
  const float* bw  = (const float*)d_in[8];
  const float* Wg1 = (const float*)d_in[9];
  const float* bg1 = (const float*)d_in[10];
  const float* Wq  = (const float*)d_in[11];
  const float* bq  = (const float*)d_in[12];
  const float* Wk  = (const float*)d_in[13];
  const float* bk  = (const float*)d_in[14];
  const float* Wv  = (const float*)d_in[15];
  const float* bv  = (const float*)d_in[16];
  const float* Wc  = (const float*)d_in[17];
  const float* bc  = (const float*)d_in[18];
  const float* gam = (const float*)d_in[19];
  const float* Wg2 = (const float*)d_in[20];
  const float* bg2 = (const float*)d_in[21];

  float* ws   = (float*)d_ws;
  float* h    = ws + OFF_H;
  float* sq   = ws + OFF_SQ;
  int*   idx  = (int*)(ws + OFF_IDX);
  float* h1   = ws + OFF_H1;
  float* wn   = ws + OFF_WN;
  float* dm   = ws + OFF_DM;
  float* coef = ws + OFF_COEF;
  float* wsum = ws + OFF_WSUM;
  float* T1   = ws + OFF_T1;
  float* h2   = ws + OFF_H2;
  float* qb   = ws + OFF_Q;
  float* kb   = ws + OFF_KK;
  float* vb   = ws + OFF_V;
  float* ao   = ws + OFF_AO;
  float* h3   = ws + OFF_H3;
  float* T2   = ws + OFF_T2;
  float* h4   = (float*)d_out;

  k_embed<<<B_*N_, 128, 0, stream>>>(x, W0, b0, h, sq);
  k_knn  <<<B_*NT_, 32, 0, stream>>>(h, sq, idx);
  k_sce  <<<B_*NT_, 256, SCE_LDS_BYTES, stream>>>(h, idx, W1, b1, W2, b2, h1);
  k_wlog <<<(B_*N_)/256, 256, 0, stream>>>(h1, Ww, bw, wn);
  k_wsum <<<B_, 256, 0, stream>>>(wn, wsum);
  k_coef <<<(B_*N_)/256, 256, 0, stream>>>(wn, wsum, dm, coef);
  // GCN: h2 = relu(Wg1 . (L h1) + bg1)
  k_T       <<<B_*C_, 256, 0, stream>>>(coef, h1, T1);
  k_gcn_conv<<<B_*NT_*8, 32, 0, stream>>>(Wg1, bg1, h1, wn, dm, T1, h2, nullptr);
  // CCI: q,k,v projections
  k_conv<<<B_*NT_*2, 32, 0, stream>>>(Wq, bq, h2, qb, nullptr, nullptr, 2);
  k_conv<<<B_*NT_*2, 32, 0, stream>>>(Wk, bk, h2, kb, nullptr, nullptr, 2);
  k_conv<<<B_*NT_*8, 32, 0, stream>>>(Wv, bv, h2, vb, nullptr, nullptr, 8);
  k_attn<<<B_*NT_, 32, 0, stream>>>(qb, kb, vb, ao);
  // h3 = h2 + gamma * relu(Wc . o + bc)
  k_conv<<<B_*NT_*8, 32, 0, stream>>>(Wc, bc, ao, h3, h2, gam, 8);
  // GS: h4 = h3 + relu(Wg2 . (L h3) + bg2)
  k_T       <<<B_*C_, 256, 0, stream>>>(coef, h3, T2);
  k_gcn_conv<<<B_*NT_*8, 32, 0, stream>>>(Wg2, bg2, h3, wn, dm, T2, h4, h3);
}